// nlwt_catone_65120294142329
// MI455X (gfx1250) — compile-verified
//
#include <hip/hip_runtime.h>

// ---------------------------------------------------------------------------
// NLWT fused forward for x: [8, 32, 512, 512] f32 -> out: [8, 128, 256, 256].
// Fused as a 4-out, 4x4-tap, stride-2 circular conv, executed with
// V_WMMA_F32_16X16X4_F32 (fp32 WMMA, K=4) on wave32.
// Staging: global -> LDS via GLOBAL_LOAD_ASYNC_TO_LDS_B128 (ASYNCcnt path).
// ---------------------------------------------------------------------------

typedef __attribute__((ext_vector_type(2))) float v2f;
typedef __attribute__((ext_vector_type(8))) float v8f;

// Types for the async-to-LDS builtin, per hipcc's own diagnostic:
//   param0: 'int __attribute__((vector_size(16))) __device__ *'  (AS1)
//   param1: expected mirror in LDS (AS3)
typedef int v4i __attribute__((vector_size(4 * sizeof(int))));
typedef v4i __attribute__((address_space(1))) v4i_g;   // global
typedef v4i __attribute__((address_space(3))) v4i_l;   // LDS

__constant__ float cM1[16] = {
     0.8664f,  0.1026f,  0.4852f, -0.0574f,
    -0.1026f,  0.8664f, -0.0574f, -0.4852f,
     0.4852f,  0.0574f, -0.8664f,  0.1026f,
     0.0574f, -0.4852f, -0.1026f, -0.8664f};

__constant__ float cM2[16] = {
     1.3968f,  0.2212f, -0.2212f, -1.3968f,
    -0.2212f,  1.3968f, -1.3968f,  0.2212f,
    -0.5412f, -1.3066f, -1.3066f, -0.5412f,
     1.3066f, -0.5412f, -0.5412f,  1.3066f};

// Fused weight W[k][dy*4+dx] = M2[k][i] * M1[i][j];
// i selects the 2x2 quadrant ((0,0)->0,(1,0)->1,(0,1)->2,(1,1)->3),
// j the position inside the 2x2 patch in (y,x)-order.
__device__ __forceinline__ float wcoef(int k, int dy, int dx) {
    const int i = (dy >> 1) + ((dx >> 1) << 1);
    const int j = ((dy & 1) << 1) + (dx & 1);
    return cM2[k * 4 + i] * cM1[i * 4 + j];
}

constexpr int BATCH = 8;
constexpr int CH    = 32;
constexpr int H     = 512;
constexpr int W     = 512;
constexpr int H1    = H / 2;   // 256
constexpr int W1    = W / 2;   // 256

__global__ __launch_bounds__(256)
void nlwt_wmma_kernel(const float* __restrict__ x, float* __restrict__ out) {
    __shared__ float lds[4][W];                 // 8 KB: the 4 input rows this block needs

    const int tid = threadIdx.x;                // 0..255  (8 wave32s)
    const int bid = blockIdx.x;                 // plane-major: good L2 reuse of row pairs
    const int y   = bid & (H1 - 1);             // output row 0..255
    const int bc  = bid >> 8;                   // plane index = b*CH + ch

    const float* plane = x + (size_t)bc * ((size_t)H * W);

    // ---- Stage wrapped input rows 2y..2y+3 into LDS (128b per lane) ----
    for (int i = tid; i < (4 * W) / 4; i += 256) {       // 512 float4s
        const int r  = i >> 7;                           // W/4 = 128 float4 per row
        const int c4 = i & 127;
        const int gr = (2 * y + r) & (H - 1);            // circular wrap in h
        const float4* src = reinterpret_cast<const float4*>(plane + (size_t)gr * W) + c4;
        float4*       dst = reinterpret_cast<float4*>(&lds[r][0]) + c4;
#if __has_builtin(__builtin_amdgcn_global_load_async_to_lds_b128)
        // CDNA5 async copy: copies 16 bytes from global address src to LDS address dst, tracked by ASYNCcnt,
        // no VGPR round-trip.
        __builtin_amdgcn_global_load_async_to_lds_b128(
            (v4i_g*)src, (v4i_l*)dst, /*offset=*/0, /*cpol=*/0);
#else
        *dst = *src;
#endif
    }
#if __has_builtin(__builtin_amdgcn_s_wait_asynccnt)
    __builtin_amdgcn_s_wait_asynccnt(0);
#else
    asm volatile("s_wait_asynccnt 0x0" ::: "memory");    // no-op if no async ops issued
#endif
    __syncthreads();

    const int lane = tid & 31;
    const int wave = tid >> 5;
    const int row  = lane & 15;                 // M-row (A) / N-column (B,D) this lane owns
    const int kb   = (lane >> 4) << 1;          // K-pair: lanes 0-15 -> K{0,1}, 16-31 -> K{2,3}

    // ---- A operand per dy-chunk: rows 0..3 = W[:, 4*dy .. 4*dy+3], rest 0 ----
    v2f A[4];
    #pragma unroll
    for (int dy = 0; dy < 4; ++dy) {
        float ax = 0.0f, ay = 0.0f;
        if (row < 4) {
            ax = wcoef(row, dy, kb);
            ay = wcoef(row, dy, kb + 1);
        }
        A[dy].x = ax;
        A[dy].y = ay;
    }

    const int b_  = bc / CH;
    const int ch  = bc % CH;
    const size_t sstr  = (size_t)CH * H1 * W1;                // subband stride in out
    const size_t base0 = (((size_t)(b_ * 4 * CH + ch)) * H1 + y) * W1;

    // ---- Each wave covers 32 output pixels of this row: 2 groups of 16 ----
    #pragma unroll
    for (int g = 0; g < 2; ++g) {
        const int xpix = wave * 32 + g * 16 + row;            // this lane's pixel column
        int col0 = 2 * xpix + kb;                             // input col of tap dx=kb
        if (col0 >= W) col0 -= W;                             // circular wrap in w (even => pair stays valid)

        v8f acc = {};
        #pragma unroll
        for (int dy = 0; dy < 4; ++dy) {
            // B(4x16): B[k][n] = in[(2y+dy) mod H][(2x_n + k) mod W]; this lane
            // supplies K = kb,kb+1 for pixel n=row: one aligned float2 from LDS.
            const v2f b = *reinterpret_cast<const v2f*>(&lds[dy][col0]);
            acc = __builtin_amdgcn_wmma_f32_16x16x4_f32(
                /*neg_a=*/false, A[dy], /*neg_b=*/false, b,
                /*c_mod=*/(short)0, acc, /*reuse_a=*/false, /*reuse_b=*/false);
        }

        // D rows 0..3 (the 4 subbands) sit in acc[0..3] of lanes 0..15.
        if (lane < 16) {
            const size_t base = base0 + (size_t)xpix;
            __builtin_nontemporal_store(acc[0], &out[base]);            // A subband
            __builtin_nontemporal_store(acc[1], &out[base + sstr]);     // B subband
            __builtin_nontemporal_store(acc[2], &out[base + 2*sstr]);   // C subband
            __builtin_nontemporal_store(acc[3], &out[base + 3*sstr]);   // D subband
        }
    }
}

extern "C" void kernel_launch(void* const* d_in, const int* in_sizes, int n_in,
                              void* d_out, int out_size, void* d_ws, size_t ws_size,
                              hipStream_t stream) {
    (void)in_sizes; (void)n_in; (void)out_size; (void)d_ws; (void)ws_size;
    const float* x = (const float*)d_in[0];
    float* out     = (float*)d_out;

    // One block per (plane, output row): 8*32*256 = 65536 blocks of 256 threads.
    const int grid = BATCH * CH * H1;
    nlwt_wmma_kernel<<<grid, 256, 0, stream>>>(x, out);
}